// CoordiPool_79044578115934
// MI455X (gfx1250) — compile-verified
//
#include <hip/hip_runtime.h>

// ---------------------------------------------------------------------------
// CoordiPool pipeline for MI455X (gfx1250, wave32)
// B=32 graphs, N=1024 nodes, F=128 feats, C=16 clusters, DK=128, E=524288
// All matrix ops use V_WMMA_F32_16X16X4_F32 (fp32 precision preserved).
// ---------------------------------------------------------------------------

#define B_    32
#define N_    1024
#define F_    128
#define C_    16
#define DK_   128
#define DEG_  16
#define E_    (B_ * N_ * DEG_)      // 524288
#define ROWS_ (B_ * N_)             // 32768
#define EPS_  1e-5f

typedef float v2f __attribute__((ext_vector_type(2)));
typedef float v8f __attribute__((ext_vector_type(8)));

// fp32 WMMA: D(16x16) = A(16x4) * B(4x16) + C
// Lane layout (ISA 7.12.2): lane L holds A row (L&15), k-pair 2*(L>>4);
// B col (L&15), same k-pair. C/D: VGPR j -> row j + 8*(L>>4), col L&15.
__device__ __forceinline__ v8f wmma_f32(v2f a, v2f b, v8f c) {
  return __builtin_amdgcn_wmma_f32_16x16x4_f32(false, a, false, b, (short)0, c,
                                               false, false);
}

// ---------------------------------------------------------------------------
// K1: edge scatter-add.  agg[src,:] += x[dst,:];  deg[src] += 1
// Block = 256 threads = 2 edges x 128 features. Coalesced 512B row reads.
// ---------------------------------------------------------------------------
__global__ void k_edge_scatter(const float* __restrict__ x,
                               const int* __restrict__ ei,
                               float* __restrict__ agg,
                               float* __restrict__ deg) {
  int e = blockIdx.x * 2 + (threadIdx.x >> 7);
  int f = threadIdx.x & 127;
  int src = ei[e];
  int dst = ei[E_ + e];
  unsafeAtomicAdd(&agg[src * F_ + f], x[dst * F_ + f]);
  if (f == 0) unsafeAtomicAdd(&deg[src], 1.0f);
}

// ---------------------------------------------------------------------------
// K2: s[row, c] = (agg[row,:]/max(deg,1)) @ W_rel[c,:] + b_rel[c]
//               +  x[row,:]             @ W_root[c,:]
// One wave per 16-row tile, C=16 covered by one WMMA tile, K=128 in steps of 4.
// ---------------------------------------------------------------------------
__global__ void k_sage(const float* __restrict__ agg,
                       const float* __restrict__ deg,
                       const float* __restrict__ x,
                       const float* __restrict__ Wrel,
                       const float* __restrict__ brel,
                       const float* __restrict__ Wroot,
                       float* __restrict__ s) {
  int wid  = (blockIdx.x * blockDim.x + threadIdx.x) >> 5;   // 0..2047
  int lane = threadIdx.x & 31;
  int m  = lane & 15;      // A row within tile == B column (channel c)
  int hi = lane >> 4;
  int kb = hi * 2;         // k-pair offset
  int r0 = wid * 16;
  int row = r0 + m;

  float invd = 1.0f / fmaxf(deg[row], 1.0f);
  const float* arow   = agg   + row * F_;
  const float* xrow   = x     + row * F_;
  const float* wrelC  = Wrel  + m * F_;   // B[k][n] = W_rel[n*F + k]
  const float* wrootC = Wroot + m * F_;

  v8f acc = {};
  for (int k = 0; k < F_; k += 4) {
    v2f a1; a1.x = arow[k + kb] * invd; a1.y = arow[k + kb + 1] * invd;
    v2f b1; b1.x = wrelC[k + kb];       b1.y = wrelC[k + kb + 1];
    acc = wmma_f32(a1, b1, acc);
    v2f a2; a2.x = xrow[k + kb];        a2.y = xrow[k + kb + 1];
    v2f b2; b2.x = wrootC[k + kb];      b2.y = wrootC[k + kb + 1];
    acc = wmma_f32(a2, b2, acc);
  }

  float bias = brel[m];
#pragma unroll
  for (int j = 0; j < 8; ++j) {
    int orow = r0 + j + 8 * hi;
    s[orow * C_ + m] = acc[j] + bias;    // 64B coalesced per half-wave
  }
}

// ---------------------------------------------------------------------------
// K3: per-channel sum and sum-of-squares over 32768 rows (BatchNorm stats).
// stats[0..15] = sum, stats[16..31] = sumsq
// ---------------------------------------------------------------------------
__global__ void k_stats(const float* __restrict__ s, float* __restrict__ stats) {
  int ch = threadIdx.x & 15;
  int r  = (blockIdx.x * blockDim.x + threadIdx.x) >> 4;
  int rs = (gridDim.x * blockDim.x) >> 4;
  float sm = 0.f, sq = 0.f;
  for (; r < ROWS_; r += rs) {
    float v = s[r * C_ + ch];
    sm += v; sq += v * v;
  }
  __shared__ float ls[32];
  if (threadIdx.x < 32) ls[threadIdx.x] = 0.f;
  __syncthreads();
  atomicAdd(&ls[ch], sm);          // ds_add_f32
  atomicAdd(&ls[16 + ch], sq);
  __syncthreads();
  if (threadIdx.x < 32) unsafeAtomicAdd(&stats[threadIdx.x], ls[threadIdx.x]);
}

// ---------------------------------------------------------------------------
// K4: BatchNorm (batch stats) + ReLU + softmax over C=16, in place.
// One thread per row (16 contiguous floats = 64B).
// ---------------------------------------------------------------------------
__global__ void k_norm_softmax(float* __restrict__ s,
                               const float* __restrict__ stats,
                               const float* __restrict__ gamma,
                               const float* __restrict__ beta) {
  int row = blockIdx.x * blockDim.x + threadIdx.x;
  if (row >= ROWS_) return;
  const float invn = 1.0f / (float)ROWS_;
  float v[C_];
  float mx = -1e30f;
#pragma unroll
  for (int c = 0; c < C_; ++c) {
    float mean = stats[c] * invn;
    float var  = stats[16 + c] * invn - mean * mean;
    float t = (s[row * C_ + c] - mean) * __frsqrt_rn(var + EPS_) * gamma[c]
              + beta[c];
    t = fmaxf(t, 0.0f);
    v[c] = t;
    mx = fmaxf(mx, t);
  }
  float ssum = 0.f;
#pragma unroll
  for (int c = 0; c < C_; ++c) { v[c] = __expf(v[c] - mx); ssum += v[c]; }
  float rcp = 1.0f / ssum;
#pragma unroll
  for (int c = 0; c < C_; ++c) s[row * C_ + c] = v[c] * rcp;
}

// ---------------------------------------------------------------------------
// K5: diff-pool  h[b,c,f] = sum_n s[b,n,c] * x[b,n,f]
// One wave per (graph, 16-wide f tile): M=C=16, N=16, K=N_=1024 step 4.
// A loads (s rows, 64B) and B loads (x rows, 64B) are coalesced.
// ---------------------------------------------------------------------------
__global__ void k_pool(const float* __restrict__ s,
                       const float* __restrict__ x,
                       float* __restrict__ h) {
  int wid  = (blockIdx.x * blockDim.x + threadIdx.x) >> 5;   // 0..255
  int lane = threadIdx.x & 31;
  int b  = wid >> 3;
  int f0 = (wid & 7) * 16;
  int c  = lane & 15;    // A row (cluster) == B column (f offset)
  int hi = lane >> 4;
  int kb = hi * 2;
  const float* sb = s + b * N_ * C_;
  const float* xb = x + b * N_ * F_;

  v8f acc = {};
  for (int k = 0; k < N_; k += 4) {
    v2f a;  a.x  = sb[(k + kb) * C_ + c];      a.y  = sb[(k + kb + 1) * C_ + c];
    v2f bb; bb.x = xb[(k + kb) * F_ + f0 + c]; bb.y = xb[(k + kb + 1) * F_ + f0 + c];
    acc = wmma_f32(a, bb, acc);
  }
#pragma unroll
  for (int j = 0; j < 8; ++j) {
    int m = j + 8 * hi;                        // cluster
    h[(b * C_ + m) * F_ + f0 + c] = acc[j];
  }
}

// ---------------------------------------------------------------------------
// K6: metal attention per graph. One block (8 waves) per graph.
// K = h@W_k^T and V = h@W_v^T via WMMA (8 dk-tiles, one per wave),
// then tiny softmax over C=16 and out = relu(attn @ V).
// hbuf padded to stride 129 -> conflict-free A-operand reads.
// ---------------------------------------------------------------------------
__global__ void k_attn(const float* __restrict__ h,
                       const float* __restrict__ metal,
                       const float* __restrict__ Wq,
                       const float* __restrict__ Wk,
                       const float* __restrict__ Wv,
                       float* __restrict__ out) {
  __shared__ float hbuf[C_ * 129];
  __shared__ float qbuf[DK_];
  __shared__ float kbuf[C_ * DK_];
  __shared__ float vbuf[C_ * DK_];
  __shared__ float sc[C_];
  __shared__ float attn[C_];

  int b = blockIdx.x;
  int t = threadIdx.x;

  for (int i = t; i < C_ * F_; i += 256)
    hbuf[(i >> 7) * 129 + (i & 127)] = h[b * C_ * F_ + i];

  if (t < DK_) {                      // Q = metal @ W_q^T  (1 x DK)
    const float* mrow = metal + b * F_;
    const float* wrow = Wq + t * F_;
    float acc = 0.f;
    for (int f = 0; f < F_; ++f) acc += mrow[f] * wrow[f];
    qbuf[t] = acc;
  }
  __syncthreads();

  {                                    // K, V tiles: wave wv -> dk tile
    int wv = t >> 5, lane = t & 31;
    int n  = lane & 15;               // A row (cluster) == B column (dk off)
    int hi = lane >> 4;
    int kb = hi * 2;
    int dk0 = wv * 16;
    const float* wkc = Wk + (dk0 + n) * F_;   // B[k][n] = W_k[(dk0+n)*F + k]
    const float* wvc = Wv + (dk0 + n) * F_;
    v8f acck = {}, accv = {};
    for (int k = 0; k < F_; k += 4) {
      v2f a;  a.x  = hbuf[n * 129 + k + kb]; a.y  = hbuf[n * 129 + k + kb + 1];
      v2f bk; bk.x = wkc[k + kb];            bk.y = wkc[k + kb + 1];
      acck = wmma_f32(a, bk, acck);
      v2f bv; bv.x = wvc[k + kb];            bv.y = wvc[k + kb + 1];
      accv = wmma_f32(a, bv, accv);
    }
#pragma unroll
    for (int j = 0; j < 8; ++j) {
      int m = j + 8 * hi;                    // cluster
      kbuf[m * DK_ + dk0 + n] = acck[j];
      vbuf[m * DK_ + dk0 + n] = accv[j];
    }
  }
  __syncthreads();

  if (t < C_) {                        // scores = Q.K^T / sqrt(DK)
    float acc = 0.f;
    for (int d = 0; d < DK_; ++d) acc += qbuf[d] * kbuf[t * DK_ + d];
    sc[t] = acc * 0.08838834764831845f;      // 1/sqrt(128)
  }
  __syncthreads();
  if (t == 0) {                        // softmax over 16 clusters
    float mx = sc[0];
    for (int c = 1; c < C_; ++c) mx = fmaxf(mx, sc[c]);
    float ssum = 0.f;
    float e[C_];
    for (int c = 0; c < C_; ++c) { e[c] = __expf(sc[c] - mx); ssum += e[c]; }
    float rcp = 1.0f / ssum;
    for (int c = 0; c < C_; ++c) attn[c] = e[c] * rcp;
  }
  __syncthreads();
  if (t < DK_) {                       // out = relu(attn @ V)
    float acc = 0.f;
#pragma unroll
    for (int c = 0; c < C_; ++c) acc += attn[c] * vbuf[c * DK_ + t];
    out[b * DK_ + t] = fmaxf(acc, 0.0f);
  }
}

// ---------------------------------------------------------------------------
// Host launcher.  Workspace layout (bytes):
//   [0,           16777216)  agg   [ROWS, F] f32
//   [16777216,    16908288)  deg   [ROWS]    f32
//   [16908288,    16908544)  stats [32]      f32 (sum | sumsq)
//   [16908544,    19005696)  s     [ROWS, C] f32
//   [19005696,    19267840)  h     [B, C, F] f32
// ---------------------------------------------------------------------------
extern "C" void kernel_launch(void* const* d_in, const int* in_sizes, int n_in,
                              void* d_out, int out_size, void* d_ws,
                              size_t ws_size, hipStream_t stream) {
  const float* x     = (const float*)d_in[0];
  const float* metal = (const float*)d_in[1];
  /* d_in[2] = batch (unused: graphs are equal-sized) */
  const int*   ei    = (const int*)d_in[3];
  const float* Wrel  = (const float*)d_in[4];
  const float* brel  = (const float*)d_in[5];
  const float* Wroot = (const float*)d_in[6];
  const float* gamma = (const float*)d_in[7];
  const float* beta  = (const float*)d_in[8];
  const float* Wq    = (const float*)d_in[9];
  const float* Wk    = (const float*)d_in[10];
  const float* Wv    = (const float*)d_in[11];
  float* out = (float*)d_out;

  char* ws = (char*)d_ws;
  float* agg   = (float*)(ws);
  float* deg   = (float*)(ws + 16777216);
  float* stats = (float*)(ws + 16777216 + 131072);
  float* s     = (float*)(ws + 16777216 + 131072 + 256);
  float* h     = (float*)(ws + 16777216 + 131072 + 256 + 2097152);

  hipMemsetAsync(ws, 0, 16777216 + 131072 + 256, stream);

  k_edge_scatter<<<E_ / 2, 256, 0, stream>>>(x, ei, agg, deg);
  k_sage<<<(ROWS_ / 16) / 4, 128, 0, stream>>>(agg, deg, x, Wrel, brel, Wroot, s);
  k_stats<<<64, 256, 0, stream>>>(s, stats);
  k_norm_softmax<<<ROWS_ / 256, 256, 0, stream>>>(s, stats, gamma, beta);
  k_pool<<<64, 128, 0, stream>>>(s, x, h);
  k_attn<<<B_, 256, 0, stream>>>(h, metal, Wq, Wk, Wv, out);
}